// SymmetryDistanceError_14551349199339
// MI455X (gfx1250) — compile-verified
//
#include <hip/hip_runtime.h>

typedef __attribute__((ext_vector_type(2))) float v2f;
typedef __attribute__((ext_vector_type(8))) float v8f;

#define NPTS 4096
#define NM   8
#define NTILE (NPTS / 16)          // 256 column/row tiles
#define NTASK (2 * NM * NTILE)     // 4096 (dir, m, row-tile) tasks

// ---------------------------------------------------------------------------
// Phase 1: gather cur = pos[batch[m]], reflect through plane m, pack as
// float4 (x, y, z, |p|^2) for both point sets.
// ---------------------------------------------------------------------------
__global__ void sde_setup(const float* __restrict__ pos,
                          const float* __restrict__ plane,
                          const int*   __restrict__ batch,
                          float* __restrict__ Pref,
                          float* __restrict__ Pcur) {
  int idx = blockIdx.x * blockDim.x + threadIdx.x;
  if (idx >= NM * NPTS) return;
  int m = idx >> 12;            // / 4096
  int n = idx & (NPTS - 1);
  int b = batch[m];
  const float* p = pos + ((size_t)b * NPTS + n) * 3;
  float cx = p[0], cy = p[1], cz = p[2];
  float nx = plane[m * 4 + 0], ny = plane[m * 4 + 1];
  float nz = plane[m * 4 + 2], dd = plane[m * 4 + 3];
  // reference uses the raw (unnormalized) plane normal
  float proj = cx * nx + cy * ny + cz * nz + dd;
  float rx = cx - 2.0f * proj * nx;
  float ry = cy - 2.0f * proj * ny;
  float rz = cz - 2.0f * proj * nz;
  float cs = cx * cx + cy * cy + cz * cz;
  float rs = rx * rx + ry * ry + rz * rz;
  *(float4*)(Pcur + (size_t)idx * 4) = make_float4(cx, cy, cz, cs);
  *(float4*)(Pref + (size_t)idx * 4) = make_float4(rx, ry, rz, rs);
}

// ---------------------------------------------------------------------------
// Phase 2: one wave per (direction, plane, row-tile).  Inner loop over all
// 256 column tiles:  D = WMMA_f32_16x16x4( [-2a | 1] , [b | |b|^2] , 0 )
// gives  -2 a.b + |b|^2  per 16x16 tile; running min in 8 acc VGPRs.
// Row norms |a|^2 and the max(.,0) clamp are applied after the min
// (both commute with min_j).  Cross-lane min via shfl_xor within each
// 16-lane half (C/D layout: lanes 0-15 -> rows v, lanes 16-31 -> rows v+8).
// ---------------------------------------------------------------------------
__global__ void sde_chamfer(const float* __restrict__ Pref,
                            const float* __restrict__ Pcur,
                            float* __restrict__ partial) {
  int wave = threadIdx.x >> 5;
  int lane = threadIdx.x & 31;
  int task = blockIdx.x * (blockDim.x >> 5) + wave;
  if (task >= NTASK) return;                 // uniform per wave

  int dir  = task >> 11;                     // / 2048
  int m    = (task >> 8) & 7;
  int tile = task & 255;

  const float* Pa = (dir == 0 ? Pref : Pcur) + (size_t)m * NPTS * 4;
  const float* Pb = (dir == 0 ? Pcur : Pref) + (size_t)m * NPTS * 4;

  int col  = lane & 15;
  int half = lane >> 4;

  // A-matrix 16x4 layout: lanes 0-15 hold K=0,1; lanes 16-31 hold K=2,3
  int rowA = tile * 16 + col;
  float2 pa = ((const float2*)(Pa + rowA * 4))[half];   // (x,y) or (z,|a|^2)
  v2f a;
  a.x = -2.0f * pa.x;
  a.y = half ? 1.0f : (-2.0f * pa.y);       // K=3 component of A is 1

  v8f acc;
#pragma unroll
  for (int k = 0; k < 8; ++k) acc[k] = 3.0e38f;

#pragma unroll 4
  for (int j = 0; j < NTILE; ++j) {
    int colB = j * 16 + col;
    // B-matrix 4x16: lanes 0-15 carry K=0,1 rows; lanes 16-31 carry K=2,3
    float2 pb = ((const float2*)(Pb + colB * 4))[half]; // (x,y) or (z,|b|^2)
    __builtin_prefetch(Pb + ((j + 4) & 255) * 16 * 4 + col * 4, 0, 1);
    v2f bm; bm.x = pb.x; bm.y = pb.y;
    v8f c = {};
    v8f d = __builtin_amdgcn_wmma_f32_16x16x4_f32(
        /*neg_a=*/false, a, /*neg_b=*/false, bm,
        /*c_mod=*/(short)0, c, /*reuse_a=*/false, /*reuse_b=*/false);
#pragma unroll
    for (int k = 0; k < 8; ++k) acc[k] = fminf(acc[k], d[k]);
  }

  // min over the 16 columns: reduce across lanes inside each 16-lane half
#pragma unroll
  for (int mask = 1; mask <= 8; mask <<= 1) {
#pragma unroll
    for (int k = 0; k < 8; ++k)
      acc[k] = fminf(acc[k], __shfl_xor(acc[k], mask, 32));
  }

  // rows owned by this half: M = v + 8*half ; add |a_M|^2, clamp, sum
  float s = 0.0f;
#pragma unroll
  for (int v = 0; v < 8; ++v) {
    int row = tile * 16 + 8 * half + v;
    float nrm = Pa[row * 4 + 3];
    s += fmaxf(acc[v] + nrm, 0.0f);
  }
  s += __shfl_xor(s, 16, 32);                // combine the two halves
  if (lane == 0) partial[task] = s;
}

// ---------------------------------------------------------------------------
// Phase 3: sum 4096 tile partials; mean = sum / (8 * 4096)
//   (cham[m] = S0[m]/4096 + S1[m]/4096 ; out = mean_m cham[m])
// ---------------------------------------------------------------------------
__global__ void sde_finalize(const float* __restrict__ partial,
                             float* __restrict__ out) {
  __shared__ float sdata[256];
  int t = threadIdx.x;
  float s = 0.0f;
  for (int i = t; i < NTASK; i += 256) s += partial[i];
  sdata[t] = s;
  __syncthreads();
  for (int off = 128; off > 0; off >>= 1) {
    if (t < off) sdata[t] += sdata[t + off];
    __syncthreads();
  }
  if (t == 0) out[0] = sdata[0] / (float)(NM * NPTS);
}

extern "C" void kernel_launch(void* const* d_in, const int* in_sizes, int n_in,
                              void* d_out, int out_size, void* d_ws, size_t ws_size,
                              hipStream_t stream) {
  const float* pos   = (const float*)d_in[0];   // (4, 4096, 3) f32
  const float* plane = (const float*)d_in[1];   // (8, 4) f32
  const int*   batch = (const int*)d_in[2];     // (8,) int

  float* ws      = (float*)d_ws;
  float* Pref    = ws;                          // 8*4096*4 floats
  float* Pcur    = Pref + (size_t)NM * NPTS * 4;
  float* partial = Pcur + (size_t)NM * NPTS * 4; // 4096 floats

  sde_setup<<<(NM * NPTS + 255) / 256, 256, 0, stream>>>(pos, plane, batch,
                                                         Pref, Pcur);
  // 4096 waves = 1024 blocks x 4 waves (128 threads, wave32)
  sde_chamfer<<<NTASK / 4, 128, 0, stream>>>(Pref, Pcur, partial);
  sde_finalize<<<1, 256, 0, stream>>>(partial, (float*)d_out);
}